// DLinear_73770358276531
// MI455X (gfx1250) — compile-verified
//
#include <hip/hip_runtime.h>

typedef __attribute__((ext_vector_type(2))) float v2f;
typedef __attribute__((ext_vector_type(8))) float v8f;

#define B_   64
#define C_   3
#define N_   512
#define L_   512
#define P_   336
#define PAD_ 12
#define KWIN_ 25

// ---------------------------------------------------------------------------
// Kernel 1: fold moving-average + seasonal/trend linears into one effective
// weight matrix Weff[P,L] and per-column bias constP[P].
//   Weff[p,l'] = Ws[p,l'] + sum_{l in [l'-12, l'+12] ∩ [0,511]} D[p,l]
//                + (l'==0   ? sum_{j=0}^{12}   (12-j)  * D[p,j] : 0)
//                + (l'==511 ? sum_{j=499}^{511}(j-499) * D[p,j] : 0)
//   with D = (Wt - Ws)/25.   constP[p] = (bs[p]+bt[p])*(Wf0+Wf1+Wf2) + bf.
// ---------------------------------------------------------------------------
__global__ void __launch_bounds__(512)
dlin_prep(const float* __restrict__ Ws, const float* __restrict__ bs,
          const float* __restrict__ Wt, const float* __restrict__ bt,
          const float* __restrict__ Wf, const float* __restrict__ bf,
          float* __restrict__ Weff, float* __restrict__ constP) {
  __shared__ float sD[L_];
  const int p = blockIdx.x;
  const int l = threadIdx.x;
  const float ws_v = Ws[p * L_ + l];
  sD[l] = (Wt[p * L_ + l] - ws_v) * (1.0f / (float)KWIN_);
  __syncthreads();

  float sum = ws_v;
  const int j0 = (l - PAD_ < 0) ? 0 : (l - PAD_);
  const int j1 = (l + PAD_ > L_ - 1) ? (L_ - 1) : (l + PAD_);
  for (int j = j0; j <= j1; ++j) sum += sD[j];
  if (l == 0) {
    for (int j = 0; j <= PAD_; ++j) sum += (float)(PAD_ - j) * sD[j];
  }
  if (l == L_ - 1) {
    for (int j = L_ - 1 - PAD_; j <= L_ - 1; ++j)
      sum += (float)(j - (L_ - 1 - PAD_)) * sD[j];
  }
  Weff[p * L_ + l] = sum;

  if (l == 0) {
    constP[p] = (bs[p] + bt[p]) * (Wf[0] + Wf[1] + Wf[2]) + bf[0];
  }
}

// ---------------------------------------------------------------------------
// Kernel 2: fused channel-fold + GEMM using V_WMMA_F32_16X16X4_F32.
// M = B*N = 32768 rows, K = 512, Ncols = 336.
// Each wave owns one 16-row tile and 112 output columns (7 WMMA accumulators,
// 336 = 3 groups * 112). Column groups are on blockIdx.x (fastest-varying in
// dispatch order) so the 3 blocks sharing an x row-stripe run back-to-back
// and x re-reads hit L2 -> HBM traffic stays near the 245 MB roofline floor.
//
// A fragment (16x4 f32): lane m = lane&15 is the row, half = lane>>4 selects
// K base {0,1} vs {2,3} -> one aligned float2 per lane per channel, folded
// with Wf (this IS the C->1 channel mix). B fragment mirrors A with
// N = lane&15 (B = Weff^T, i.e. rows of Weff).
// ---------------------------------------------------------------------------
__global__ void __launch_bounds__(256)
dlin_gemm(const float* __restrict__ x, const float* __restrict__ Weff,
          const float* __restrict__ constP, const float* __restrict__ Wf,
          float* __restrict__ out) {
  const int lane   = threadIdx.x & 31;
  const int wave   = threadIdx.x >> 5;
  const int lane_m = lane & 15;
  const int half   = lane >> 4;

  const int row_tile = blockIdx.y * 8 + wave;   // 2048 tiles of 16 rows
  const int row0 = row_tile * 16;
  const int row  = row0 + lane_m;               // A-row handled by this lane
  const int b = row >> 9;                       // N_ == 512
  const int n = row & (N_ - 1);

  const unsigned xbase = ((unsigned)(b * C_) * N_ + (unsigned)n) * L_;
  const int p0 = blockIdx.x * 112;              // 3 groups * 112 = 336

  const float wf0 = Wf[0], wf1 = Wf[1], wf2 = Wf[2];

  int pr[7];
#pragma unroll
  for (int t = 0; t < 7; ++t) pr[t] = (p0 + t * 16 + lane_m) * L_;

  v8f acc[7];
#pragma unroll
  for (int t = 0; t < 7; ++t) {
    const float cv = constP[p0 + t * 16 + lane_m];
#pragma unroll
    for (int j = 0; j < 8; ++j) acc[t][j] = cv;
  }

  const int klane = 2 * half;
#pragma unroll 2
  for (int k = 0; k < L_; k += 4) {
    const int kk = k + klane;
    // channel fold: y = Wf0*x0 + Wf1*x1 + Wf2*x2
    v2f a0 = *(const v2f*)(x + xbase + (unsigned)kk);
    v2f a1 = *(const v2f*)(x + xbase + (unsigned)(N_ * L_) + (unsigned)kk);
    v2f a2 = *(const v2f*)(x + xbase + (unsigned)(2 * N_ * L_) + (unsigned)kk);
    v2f a  = a0 * wf0 + a1 * wf1 + a2 * wf2;

#pragma unroll
    for (int t = 0; t < 7; ++t) {
      v2f w = *(const v2f*)(Weff + pr[t] + kk);
      acc[t] = __builtin_amdgcn_wmma_f32_16x16x4_f32(false, a, false, w,
                                                     (short)0, acc[t],
                                                     false, false);
    }
  }

  // D layout: VGPR j holds M = j + 8*half, N = lane&15
#pragma unroll
  for (int j = 0; j < 8; ++j) {
    const int m = j + 8 * half;
    float* orow = out + (size_t)(row0 + m) * P_;
#pragma unroll
    for (int t = 0; t < 7; ++t) {
      orow[p0 + t * 16 + lane_m] = acc[t][j];
    }
  }
}

// ---------------------------------------------------------------------------
extern "C" void kernel_launch(void* const* d_in, const int* in_sizes, int n_in,
                              void* d_out, int out_size, void* d_ws, size_t ws_size,
                              hipStream_t stream) {
  const float* x  = (const float*)d_in[0];
  const float* Ws = (const float*)d_in[1];
  const float* bs = (const float*)d_in[2];
  const float* Wt = (const float*)d_in[3];
  const float* bt = (const float*)d_in[4];
  const float* Wf = (const float*)d_in[5];
  const float* bf = (const float*)d_in[6];
  float* out = (float*)d_out;

  // Workspace: Weff (336*512 f32 = 688KB) followed by constP (336 f32).
  float* Weff   = (float*)d_ws;
  float* constP = Weff + (size_t)P_ * L_;

  dlin_prep<<<dim3(P_), dim3(L_), 0, stream>>>(Ws, bs, Wt, bt, Wf, bf,
                                               Weff, constP);

  // grid.x = 3 column groups (fastest-varying -> L2 reuse of x stripes),
  // grid.y = 2048 row tiles / 8 waves per block = 256.
  // Exact tiling, no divergence -> EXEC all ones at every WMMA.
  dlin_gemm<<<dim3(P_ / 112, (B_ * N_) / (16 * 8)), dim3(256), 0, stream>>>(
      x, Weff, constP, Wf, out);
}